// GAT_21174188769407
// MI455X (gfx1250) — compile-verified
//
#include <hip/hip_runtime.h>
#include <math.h>

typedef __attribute__((ext_vector_type(2))) float v2f;
typedef __attribute__((ext_vector_type(8))) float v8f;

#define N_NODES 100000
#define E_RAW   1000000
#define E_TOT   1100000
#define MAXF    64

// Monotone encoding of f32 into u32 so atomicMax(u32) implements float max.
__device__ __forceinline__ unsigned enc_f32(float f) {
    unsigned u = __float_as_uint(f);
    return (u & 0x80000000u) ? ~u : (u | 0x80000000u);
}
__device__ __forceinline__ float dec_f32(unsigned u) {
    u = (u & 0x80000000u) ? (u & 0x7FFFFFFFu) : ~u;
    return __uint_as_float(u);
}

// ---------------------------------------------------------------------------
// xl = H*Wl, xr = H*Wr via V_WMMA_F32_16X16X4_F32.
// Block = 256 threads = 8 waves; wave w computes one 16x16 output tile:
//   waves [0, nct)      -> Wl col-tiles, waves [nct, 2*nct) -> Wr col-tiles.
// W matrices staged in LDS zero-padded to 64 columns (handles Fout=40).
// ---------------------------------------------------------------------------
__global__ void __launch_bounds__(256)
gat_gemm(const float* __restrict__ H, const float* __restrict__ Wl,
         const float* __restrict__ Wr, float* __restrict__ XL,
         float* __restrict__ XR, int Fin, int Fout)
{
    __shared__ float sW[2 * 128 * 64];   // 64 KB, padded [mat][Fin][64]
    const int tid = threadIdx.x;
    const int per = Fin * 64;
    for (int idx = tid; idx < 2 * per; idx += 256) {
        int mat = idx / per;
        int rem = idx - mat * per;
        int k = rem >> 6, n = rem & 63;
        const float* W = mat ? Wr : Wl;
        sW[idx] = (n < Fout) ? W[k * Fout + n] : 0.0f;
    }
    __syncthreads();

    const int wave = tid >> 5;
    const int lane = tid & 31;
    const int nct  = (Fout + 15) >> 4;          // 4 (F=64) or 3 (F=40)
    if (wave >= 2 * nct) return;                // wave-uniform: EXEC stays all-1s
    const int mat  = (wave >= nct) ? 1 : 0;
    const int n0   = (mat ? (wave - nct) : wave) << 4;
    const int m0   = blockIdx.x << 4;           // N_NODES % 16 == 0
    const int half = lane >> 4;                 // 0: K={0,1}, 1: K={2,3}
    const int l15  = lane & 15;

    const float* __restrict__ Arow  = H + (size_t)(m0 + l15) * Fin + half * 2;
    const float* __restrict__ Bbase = sW + mat * per + n0 + l15;

    v8f c = {};
    for (int k0 = 0; k0 < Fin; k0 += 4) {
        float2 av = *(const float2*)(Arow + k0);        // A: 16x4 f32 tile
        v2f A; A.x = av.x; A.y = av.y;
        const int kb = k0 + half * 2;
        v2f B; B.x = Bbase[kb * 64]; B.y = Bbase[(kb + 1) * 64];  // B: 4x16
        c = __builtin_amdgcn_wmma_f32_16x16x4_f32(false, A, false, B,
                                                  (short)0, c, false, false);
    }

    float* __restrict__ D = mat ? XR : XL;
    const int col = n0 + l15;
    if (col < Fout) {
        #pragma unroll
        for (int v = 0; v < 8; ++v) {
            int row = m0 + v + half * 8;        // C/D layout: lanes16-31 -> M+8
            D[(size_t)row * Fout + col] = c[v];
        }
    }
}

// ---------------------------------------------------------------------------
// Zero per-layer accumulators. enc(-inf) ordering: 0 is below every encoding.
// ---------------------------------------------------------------------------
__global__ void gat_init(unsigned* __restrict__ menc, float* __restrict__ ssum,
                         float* __restrict__ outa, int Fout)
{
    int idx = blockIdx.x * 256 + threadIdx.x;
    if (idx < N_NODES) { menc[idx] = 0u; ssum[idx] = 0.0f; }
    if (idx < N_NODES * Fout) outa[idx] = 0.0f;
}

// ---------------------------------------------------------------------------
// Per-edge attention logit: e = att . leaky_relu(xl[src] + xr[dst]).
// One wave32 per edge; lanes stride the feature dim; shfl_xor reduction.
// ---------------------------------------------------------------------------
__global__ void __launch_bounds__(256)
gat_edge_score(const float* __restrict__ XL, const float* __restrict__ XR,
               const long long* __restrict__ ei, const float* __restrict__ att,
               float* __restrict__ esc, unsigned* __restrict__ menc, int Fout)
{
    int edge = (int)((blockIdx.x * 256 + threadIdx.x) >> 5);
    if (edge >= E_TOT) return;
    const int lane = threadIdx.x & 31;
    int src, dst;
    if (edge < E_RAW) { src = (int)ei[edge]; dst = (int)ei[E_RAW + edge]; }
    else              { src = dst = edge - E_RAW; }          // self loops
    float acc = 0.0f;
    for (int f = lane; f < Fout; f += 32) {
        float v = XL[(size_t)src * Fout + f] + XR[(size_t)dst * Fout + f];
        v = (v > 0.0f) ? v : 0.2f * v;                       // leaky_relu 0.2
        acc += v * att[f];
    }
    for (int o = 16; o; o >>= 1) acc += __shfl_xor(acc, o, 32);
    if (lane == 0) {
        esc[edge] = acc;
        atomicMax(menc + dst, enc_f32(acc));
    }
}

// a = exp(e - m[dst]); s[dst] += a   (in-place over esc)
__global__ void gat_edge_exp(const long long* __restrict__ ei,
                             const unsigned* __restrict__ menc,
                             float* __restrict__ esc, float* __restrict__ ssum)
{
    int edge = blockIdx.x * 256 + threadIdx.x;
    if (edge >= E_TOT) return;
    int dst = (edge < E_RAW) ? (int)ei[E_RAW + edge] : edge - E_RAW;
    float a = expf(esc[edge] - dec_f32(menc[dst]));
    esc[edge] = a;
    atomicAdd(ssum + dst, a);
}

// out[dst] += (a/s[dst]) * xl[src]   — wave per edge, atomic f32 adds (L2-resident)
__global__ void __launch_bounds__(256)
gat_edge_scatter(const float* __restrict__ XL, const float* __restrict__ esc,
                 const float* __restrict__ ssum, const long long* __restrict__ ei,
                 float* __restrict__ outa, int Fout)
{
    int edge = (int)((blockIdx.x * 256 + threadIdx.x) >> 5);
    if (edge >= E_TOT) return;
    const int lane = threadIdx.x & 31;
    int src, dst;
    if (edge < E_RAW) { src = (int)ei[edge]; dst = (int)ei[E_RAW + edge]; }
    else              { src = dst = edge - E_RAW; }
    const float alpha = esc[edge] / ssum[dst];
    for (int f = lane; f < Fout; f += 32)
        atomicAdd(outa + (size_t)dst * Fout + f, alpha * XL[(size_t)src * Fout + f]);
}

// h_next = out + bias (layers 0,1)
__global__ void gat_bias(const float* __restrict__ outa, const float* __restrict__ b,
                         float* __restrict__ hnext, int Fout)
{
    int idx = blockIdx.x * 256 + threadIdx.x;
    if (idx >= N_NODES * Fout) return;
    hnext[idx] = outa[idx] + b[idx % Fout];
}

// final layer: log_softmax(out + bias) over Fout=40; one wave per node.
__global__ void __launch_bounds__(256)
gat_final(const float* __restrict__ outa, const float* __restrict__ b,
          float* __restrict__ out, int Fout)
{
    int node = (int)((blockIdx.x * 256 + threadIdx.x) >> 5);
    if (node >= N_NODES) return;
    const int lane = threadIdx.x & 31;
    const int f1 = lane + 32;
    const bool has1 = f1 < Fout;                 // Fout=40: lanes 0..7 carry 2nd elem
    float v0 = outa[(size_t)node * Fout + lane] + b[lane];
    float v1 = has1 ? outa[(size_t)node * Fout + f1] + b[f1] : -INFINITY;
    float mx = fmaxf(v0, v1);
    for (int o = 16; o; o >>= 1) mx = fmaxf(mx, __shfl_xor(mx, o, 32));
    float se = expf(v0 - mx) + (has1 ? expf(v1 - mx) : 0.0f);
    for (int o = 16; o; o >>= 1) se += __shfl_xor(se, o, 32);
    const float lse = mx + logf(se);
    out[(size_t)node * Fout + lane] = v0 - lse;
    if (has1) out[(size_t)node * Fout + f1] = v1 - lse;
}

extern "C" void kernel_launch(void* const* d_in, const int* in_sizes, int n_in,
                              void* d_out, int out_size, void* d_ws, size_t ws_size,
                              hipStream_t stream)
{
    const float*     x  = (const float*)d_in[0];
    const long long* ei = (const long long*)d_in[1];   // edge_index is int64 [2, 1e6]

    // Workspace carve-up (~108 MB total)
    float*    xl   = (float*)d_ws;
    float*    xr   = xl   + (size_t)N_NODES * MAXF;
    float*    hbuf = xr   + (size_t)N_NODES * MAXF;
    float*    esc  = hbuf + (size_t)N_NODES * MAXF;
    unsigned* menc = (unsigned*)(esc + E_TOT);
    float*    ssum = (float*)(menc + N_NODES);
    float*    outa = ssum + N_NODES;

    const int dims[3][2] = {{128, 64}, {64, 64}, {64, 40}};
    for (int i = 0; i < 3; ++i) {
        const int Fin = dims[i][0], Fout = dims[i][1];
        const float* H   = (i == 0) ? x : hbuf;
        const float* Wl  = (const float*)d_in[2 + 4 * i];
        const float* Wr  = (const float*)d_in[3 + 4 * i];
        const float* att = (const float*)d_in[4 + 4 * i];
        const float* b   = (const float*)d_in[5 + 4 * i];

        gat_init<<<(N_NODES * Fout + 255) / 256, 256, 0, stream>>>(menc, ssum, outa, Fout);
        gat_gemm<<<N_NODES / 16, 256, 0, stream>>>(H, Wl, Wr, xl, xr, Fin, Fout);
        gat_edge_score<<<(E_TOT * 32 + 255) / 256, 256, 0, stream>>>(xl, xr, ei, att, esc, menc, Fout);
        gat_edge_exp<<<(E_TOT + 255) / 256, 256, 0, stream>>>(ei, menc, esc, ssum);
        gat_edge_scatter<<<(E_TOT * 32 + 255) / 256, 256, 0, stream>>>(xl, esc, ssum, ei, outa, Fout);
        if (i < 2)
            gat_bias<<<(N_NODES * Fout + 255) / 256, 256, 0, stream>>>(outa, b, hbuf, Fout);
        else
            gat_final<<<(N_NODES + 7) / 8, 256, 0, stream>>>(outa, b, (float*)d_out, Fout);
    }
}